// GNNEncoderWithFallback_62577673503028
// MI455X (gfx1250) — compile-verified
//
#include <hip/hip_runtime.h>
#include <hip/hip_bf16.h>

typedef __attribute__((ext_vector_type(2))) float v2f;
typedef __attribute__((ext_vector_type(8))) float v8f;

#define N_NODES  50000
#define N_EDGES  800000
#define N_GRAPHS 512
#define HIDF     128

// ---------------------------------------------------------------- utilities
__global__ void zero_f32(float* __restrict__ p, int n) {
    int t = blockIdx.x * blockDim.x + threadIdx.x;
    if (t < n) p[t] = 0.0f;
}

__global__ void deg_accum(const int* __restrict__ dst, float* __restrict__ deg, int ne) {
    int e = blockIdx.x * blockDim.x + threadIdx.x;
    if (e < ne)
        __hip_atomic_fetch_add(&deg[dst[e]], 1.0f, __ATOMIC_RELAXED, __HIP_MEMORY_SCOPE_AGENT);
}

__global__ void finalize_dis(float* __restrict__ dis, int n) {
    int t = blockIdx.x * blockDim.x + threadIdx.x;
    if (t < n) dis[t] = rsqrtf(dis[t] + 1.0f);   // +1 = self loop
}

// ------------------------------------------------------- FP32 WMMA GEMM
// C[N,128] = A[N,128] @ W[128,128].  GATHER: A row = emb[x_type[row]].
// One block = 16 output rows; wave w (of 8) owns 16-col tile w.
// ISA 32-bit fragment layouts (wave32):
//   A 16x4:  lane L -> M = L&15, VGPR0 = K=(L>>4)*2, VGPR1 = K+1
//   B 4x16:  lane L -> N = L&15, VGPR0 = K=(L>>4)*2, VGPR1 = K+1
//   C 16x16: VGPR i -> M = i + (L>>4)*8, N = L&15
template <bool GATHER>
__global__ __launch_bounds__(256) void gemm_wmma_f32(
        const float* __restrict__ Abase,  // GATHER ? emb : x
        const int*   __restrict__ xtype,
        const float* __restrict__ W,
        float*       __restrict__ out,
        int nrows) {
    const int lane  = threadIdx.x & 31;
    const int wave  = threadIdx.x >> 5;       // 0..7 -> column tile
    const int m     = lane & 15;
    const int kpair = (lane >> 4) * 2;        // 0 or 2
    const int row   = blockIdx.x * 16 + m;
    const int col   = wave * 16 + m;

    const float* arow;
    if (GATHER) arow = Abase + (size_t)xtype[row] * HIDF;
    else        arow = Abase + (size_t)row * HIDF;

    v8f acc = {};
#pragma unroll
    for (int k0 = 0; k0 < HIDF; k0 += 4) {
        v2f a, b;
        a.x = arow[k0 + kpair];
        a.y = arow[k0 + kpair + 1];
        b.x = W[(size_t)(k0 + kpair) * HIDF + col];
        b.y = W[(size_t)(k0 + kpair + 1) * HIDF + col];
        acc = __builtin_amdgcn_wmma_f32_16x16x4_f32(
                  false, a, false, b, (short)0, acc, false, false);
    }

    const int rbase = blockIdx.x * 16 + (lane >> 4) * 8;
#pragma unroll
    for (int i = 0; i < 8; ++i)
        out[(size_t)(rbase + i) * HIDF + col] = acc[i];
}

// ------------------------------------------ self-loop + bias initialization
// a[n,f] = h[n,f] * dis[n]^2 + bias[f]
__global__ void self_loop_bias(const float* __restrict__ h, const float* __restrict__ dis,
                               const float* __restrict__ bias, float* __restrict__ a, int n) {
    int t = blockIdx.x * blockDim.x + threadIdx.x;
    if (t >= n * HIDF) return;
    int node = t >> 7, f = t & (HIDF - 1);
    float d = dis[node];
    a[t] = h[t] * d * d + bias[f];
}

// ------------------------------------------------------- edge scatter-add
// one wave per edge, 4 features per lane
__global__ void edge_scatter(const int* __restrict__ src, const int* __restrict__ dst,
                             const float* __restrict__ dis, const float* __restrict__ h,
                             float* __restrict__ a, int ne) {
    int t = blockIdx.x * blockDim.x + threadIdx.x;
    int e = t >> 5, lane = t & 31;
    if (e >= ne) return;
    int s = src[e], d = dst[e];
    float norm = dis[s] * dis[d];
    const float* hr = h + (size_t)s * HIDF;
    float*       ar = a + (size_t)d * HIDF;
#pragma unroll
    for (int i = 0; i < 4; ++i) {
        int f = lane + i * 32;
        __hip_atomic_fetch_add(&ar[f], hr[f] * norm,
                               __ATOMIC_RELAXED, __HIP_MEMORY_SCOPE_AGENT);
    }
}

__global__ void relu_inplace(float* __restrict__ p, int n) {
    int t = blockIdx.x * blockDim.x + threadIdx.x;
    if (t < n) p[t] = fmaxf(p[t], 0.0f);
}

// ---------------------------------------------------------------- pooling
__global__ void pool_scatter(const float* __restrict__ x, const int* __restrict__ batch,
                             float* __restrict__ out, int nnodes) {
    int t = blockIdx.x * blockDim.x + threadIdx.x;
    int node = t >> 5, lane = t & 31;
    if (node >= nnodes) return;
    int g = batch[node];
    const float* xr = x + (size_t)node * HIDF;
    float*       orow = out + (size_t)g * HIDF;
#pragma unroll
    for (int i = 0; i < 4; ++i) {
        int f = lane + i * 32;
        __hip_atomic_fetch_add(&orow[f], xr[f],
                               __ATOMIC_RELAXED, __HIP_MEMORY_SCOPE_AGENT);
    }
}

// ---------------------------------------------------------------- driver
extern "C" void kernel_launch(void* const* d_in, const int* in_sizes, int n_in,
                              void* d_out, int out_size, void* d_ws, size_t ws_size,
                              hipStream_t stream) {
    const int*   x_type = (const int*)  d_in[0];
    const int*   eidx   = (const int*)  d_in[1];   // [2, E] row-major
    const int*   batch  = (const int*)  d_in[2];
    const float* emb    = (const float*)d_in[3];
    const float* W1     = (const float*)d_in[4];
    const float* b1     = (const float*)d_in[5];
    const float* W2     = (const float*)d_in[6];
    const float* b2     = (const float*)d_in[7];
    float* out = (float*)d_out;

    const int* src = eidx;            // edge_index[0]
    const int* dst = eidx + N_EDGES;  // edge_index[1]

    float* dis   = (float*)d_ws;                        // N floats
    float* buf_h = dis + N_NODES;                       // N*128 floats
    float* buf_a = buf_h + (size_t)N_NODES * HIDF;      // N*128 floats

    const int B = 256;
    const int gN    = (N_NODES + B - 1) / B;
    const int gE    = (N_EDGES + B - 1) / B;
    const int gNF   = (N_NODES * HIDF + B - 1) / B;
    const int gE32  = ((N_EDGES * 32) + B - 1) / B;
    const int gN32  = ((N_NODES * 32) + B - 1) / B;
    const int gOut  = (N_GRAPHS * HIDF + B - 1) / B;
    const int gGemm = N_NODES / 16;   // 3125, exact

    // degree -> dis = rsqrt(deg + 1)
    zero_f32<<<gN, B, 0, stream>>>(dis, N_NODES);
    deg_accum<<<gE, B, 0, stream>>>(dst, dis, N_EDGES);
    finalize_dis<<<gN, B, 0, stream>>>(dis, N_NODES);

    // ---- layer 1: h = emb[x_type] @ W1 ; a = scatter(h*norm) + b1 ; relu
    gemm_wmma_f32<true><<<gGemm, B, 0, stream>>>(emb, x_type, W1, buf_h, N_NODES);
    self_loop_bias<<<gNF, B, 0, stream>>>(buf_h, dis, b1, buf_a, N_NODES);
    edge_scatter<<<gE32, B, 0, stream>>>(src, dst, dis, buf_h, buf_a, N_EDGES);
    relu_inplace<<<gNF, B, 0, stream>>>(buf_a, N_NODES * HIDF);

    // ---- layer 2: h = a @ W2 ; a = scatter(h*norm) + b2
    gemm_wmma_f32<false><<<gGemm, B, 0, stream>>>(buf_a, nullptr, W2, buf_h, N_NODES);
    self_loop_bias<<<gNF, B, 0, stream>>>(buf_h, dis, b2, buf_a, N_NODES);
    edge_scatter<<<gE32, B, 0, stream>>>(src, dst, dis, buf_h, buf_a, N_EDGES);

    // ---- pooling
    zero_f32<<<gOut, B, 0, stream>>>(out, N_GRAPHS * HIDF);
    pool_scatter<<<gN32, B, 0, stream>>>(buf_a, batch, out, N_NODES);
}